// GAT_15908558865648
// MI455X (gfx1250) — compile-verified
//
#include <hip/hip_runtime.h>
#include <math.h>

#define NN 50000
#define EE_RAW 1600000
#define DD 128
#define LL 3
#define EE_TOT (EE_RAW + NN)     // edges + self loops
#define EPS 1e-5f
#define SLOPE 0.2f

typedef float v2f __attribute__((ext_vector_type(2)));
typedef float v8f __attribute__((ext_vector_type(8)));

// ---------------- utility fills ----------------
__global__ void fill_i32(int* p, int v, int n) {
    int i = blockIdx.x * blockDim.x + threadIdx.x;
    if (i < n) p[i] = v;
}
__global__ void fill_f32(float* p, float v, int n) {
    int i = blockIdx.x * blockDim.x + threadIdx.x;
    if (i < n) p[i] = v;
}

// ---------------- CSR build ----------------
__global__ void count_edges(const int* __restrict__ ei, int* __restrict__ counts) {
    int t = blockIdx.x * blockDim.x + threadIdx.x;
    if (t >= EE_TOT) return;
    int dst = (t < EE_RAW) ? ei[EE_RAW + t] : (t - EE_RAW);
    atomicAdd(&counts[dst], 1);
}

// single-block hierarchical exclusive scan over N=50000 counts
__global__ void scan_counts(const int* __restrict__ counts,
                            int* __restrict__ row_start,
                            int* __restrict__ cursor) {
    __shared__ int tsum[1024];
    int t = threadIdx.x;
    const int chunk = (NN + 1023) >> 10;            // 49
    int beg = t * chunk;
    int end = beg + chunk; if (end > NN) end = NN;
    int s = 0;
    for (int i = beg; i < end; ++i) s += counts[i];
    tsum[t] = s;
    __syncthreads();
    for (int off = 1; off < 1024; off <<= 1) {
        int v = (t >= off) ? tsum[t - off] : 0;
        __syncthreads();
        tsum[t] += v;
        __syncthreads();
    }
    int run = (t == 0) ? 0 : tsum[t - 1];           // exclusive prefix of this chunk
    for (int i = beg; i < end; ++i) {
        row_start[i] = run;
        cursor[i]    = run;
        run += counts[i];
    }
    if (t == 1023) row_start[NN] = tsum[1023];      // total = EE_TOT
}

__global__ void fill_csr(const int* __restrict__ ei,
                         int* __restrict__ cursor,
                         int* __restrict__ csr_src) {
    int t = blockIdx.x * blockDim.x + threadIdx.x;
    if (t >= EE_TOT) return;
    int src = (t < EE_RAW) ? ei[t]            : (t - EE_RAW);
    int dst = (t < EE_RAW) ? ei[EE_RAW + t]   : (t - EE_RAW);
    int slot = atomicAdd(&cursor[dst], 1);
    csr_src[slot] = src;
}

// ---------------- WMMA GEMM: h[N,128] = z[N,128] @ W[128,128] (fp32) ----------------
// one wave per 16x16 output tile; K stepped by 4 via V_WMMA_F32_16X16X4_F32
__global__ void gemm_wmma_f32(const float* __restrict__ z,
                              const float* __restrict__ W,
                              float* __restrict__ h) {
    int wave = blockIdx.x * (blockDim.x >> 5) + (threadIdx.x >> 5);
    int lane = threadIdx.x & 31;
    int row_tile = wave >> 3;                       // 8 column tiles (128/16)
    int col_tile = wave & 7;
    if (row_tile >= NN / 16) return;                // uniform per wave
    int half = lane >> 4;                           // 0: lanes 0-15, 1: lanes 16-31
    int lr   = lane & 15;
    int row = row_tile * 16 + lr;                   // A-fragment row (M)
    int col = col_tile * 16 + lr;                   // B-fragment col (N)
    const float* zrow = z + (size_t)row * DD;
    v8f acc = {};
#pragma unroll 4
    for (int k = 0; k < DD; k += 4) {
        // A 16x4: VGPR0 = K=k+2*half, VGPR1 = K=k+2*half+1 -> contiguous float2
        v2f a = *(const v2f*)(zrow + k + 2 * half);
        // B 4x16: VGPR v holds rows {k+v (lanes<16), k+v+2 (lanes>=16)}
        v2f b;
        b[0] = W[(size_t)(k + 2 * half)     * DD + col];
        b[1] = W[(size_t)(k + 2 * half + 1) * DD + col];
        acc = __builtin_amdgcn_wmma_f32_16x16x4_f32(
            false, a, false, b, (short)0, acc, false, false);
    }
    // C/D: VGPR j -> M = j + 8*half, N = lr
#pragma unroll
    for (int j = 0; j < 8; ++j)
        h[(size_t)(row_tile * 16 + j + 8 * half) * DD + col] = acc[j];
}

// ---------------- attention logits: as[n]=h[n].a_src, ad[n]=h[n].a_dst ----------------
__global__ void attn_dots(const float* __restrict__ h,
                          const float* __restrict__ a_src,
                          const float* __restrict__ a_dst,
                          float* __restrict__ as, float* __restrict__ ad) {
    int node = blockIdx.x * (blockDim.x >> 5) + (threadIdx.x >> 5);
    int lane = threadIdx.x & 31;
    if (node >= NN) return;
    const float* hp = h + (size_t)node * DD + lane * 4;
    float ps = 0.f, pd = 0.f;
#pragma unroll
    for (int j = 0; j < 4; ++j) {
        float v = hp[j];
        ps += v * a_src[lane * 4 + j];
        pd += v * a_dst[lane * 4 + j];
    }
#pragma unroll
    for (int off = 16; off >= 1; off >>= 1) {
        ps += __shfl_xor(ps, off, 32);
        pd += __shfl_xor(pd, off, 32);
    }
    if (lane == 0) { as[node] = ps; ad[node] = pd; }
}

// ---------------- per-dst online-softmax aggregation (wave per node) ----------------
__global__ void aggregate(const int* __restrict__ row_start,
                          const int* __restrict__ csr_src,
                          const float* __restrict__ as,
                          const float* __restrict__ ad,
                          const float* __restrict__ h,
                          float* __restrict__ zout) {
    int node = blockIdx.x * (blockDim.x >> 5) + (threadIdx.x >> 5);
    int lane = threadIdx.x & 31;
    if (node >= NN) return;
    int beg = row_start[node];
    int end = row_start[node + 1];
    float adn = ad[node];
    float m = -INFINITY, lsum = 0.f;
    float acc0 = 0.f, acc1 = 0.f, acc2 = 0.f, acc3 = 0.f;
    for (int e = beg; e < end; ++e) {
        int s = csr_src[e];
        float ev = as[s] + adn;
        ev = ev > 0.f ? ev : SLOPE * ev;            // leaky_relu
        float mnew = fmaxf(m, ev);
        float scale = __expf(m - mnew);             // 0 on first iter (m=-inf)
        float pe    = __expf(ev - mnew);
        lsum = lsum * scale + pe;
        const float* hp = h + (size_t)s * DD + lane * 4;
        acc0 = acc0 * scale + pe * hp[0];
        acc1 = acc1 * scale + pe * hp[1];
        acc2 = acc2 * scale + pe * hp[2];
        acc3 = acc3 * scale + pe * hp[3];
        m = mnew;
    }
    float inv = 1.f / lsum;                         // self-loop guarantees lsum>0
    float* zp = zout + (size_t)node * DD + lane * 4;
    zp[0] = acc0 * inv; zp[1] = acc1 * inv; zp[2] = acc2 * inv; zp[3] = acc3 * inv;
    // conv bias b[l] omitted: it cancels exactly inside train-mode BatchNorm.
}

// ---------------- BatchNorm ----------------
__global__ void bn_stats(const float* __restrict__ z, float* __restrict__ sums) {
    int col    = threadIdx.x & 127;
    int rowoff = threadIdx.x >> 7;                  // 0 or 1 (256 threads)
    float s = 0.f, sq = 0.f;
    for (int r = blockIdx.x * 2 + rowoff; r < NN; r += gridDim.x * 2) {
        float v = z[(size_t)r * DD + col];
        s += v; sq += v * v;
    }
    atomicAdd(&sums[col], s);
    atomicAdd(&sums[DD + col], sq);
}

__global__ void bn_apply_relu(const float* __restrict__ z,
                              const float* __restrict__ sums,
                              float* __restrict__ out) {
    int i = blockIdx.x * blockDim.x + threadIdx.x;
    if (i >= NN * DD) return;
    int col = i & 127;
    const float invN = 1.f / (float)NN;
    float mu  = sums[col] * invN;
    float var = sums[DD + col] * invN - mu * mu;
    float v = (z[i] - mu) * rsqrtf(var + EPS);
    out[i] = v > 0.f ? v : 0.f;
}

// ---------------- launch ----------------
extern "C" void kernel_launch(void* const* d_in, const int* in_sizes, int n_in,
                              void* d_out, int out_size, void* d_ws, size_t ws_size,
                              hipStream_t stream) {
    const float* x     = (const float*)d_in[0];   // [N,D]
    const int*   ei    = (const int*)  d_in[1];   // [2,E]
    const float* W     = (const float*)d_in[2];   // [L,D,D]
    const float* a_src = (const float*)d_in[3];   // [L,D]
    const float* a_dst = (const float*)d_in[4];   // [L,D]
    float* out = (float*)d_out;

    // workspace carve-up (256B aligned)
    char* ws = (char*)d_ws;
    size_t off = 0;
    auto take = [&](size_t bytes) {
        void* p = ws + off;
        off = (off + bytes + 255) & ~(size_t)255;
        return p;
    };
    float* h        = (float*)take((size_t)NN * DD * sizeof(float));
    float* zagg     = (float*)take((size_t)NN * DD * sizeof(float));
    float* as       = (float*)take((size_t)NN * sizeof(float));
    float* ad       = (float*)take((size_t)NN * sizeof(float));
    float* sums     = (float*)take(2 * DD * sizeof(float));
    int*   counts   = (int*)  take((size_t)NN * sizeof(int));
    int*   row_start= (int*)  take((size_t)(NN + 1) * sizeof(int));
    int*   cursor   = (int*)  take((size_t)NN * sizeof(int));
    int*   csr_src  = (int*)  take((size_t)EE_TOT * sizeof(int));
    (void)ws_size; (void)in_sizes; (void)n_in; (void)out_size;

    // ---- CSR build (edge structure is layer-invariant) ----
    fill_i32<<<(NN + 255) / 256, 256, 0, stream>>>(counts, 0, NN);
    count_edges<<<(EE_TOT + 255) / 256, 256, 0, stream>>>(ei, counts);
    scan_counts<<<1, 1024, 0, stream>>>(counts, row_start, cursor);
    fill_csr<<<(EE_TOT + 255) / 256, 256, 0, stream>>>(ei, cursor, csr_src);

    const int nodeWaveBlocks = (NN + 7) / 8;        // 8 waves / 256-thread block
    const float* zin = x;
    for (int l = 0; l < LL; ++l) {
        gemm_wmma_f32<<<(NN / 16) * 8 / 8, 256, 0, stream>>>(zin, W + (size_t)l * DD * DD, h);
        attn_dots<<<nodeWaveBlocks, 256, 0, stream>>>(h, a_src + l * DD, a_dst + l * DD, as, ad);
        aggregate<<<nodeWaveBlocks, 256, 0, stream>>>(row_start, csr_src, as, ad, h, zagg);
        fill_f32<<<1, 256, 0, stream>>>(sums, 0.f, 2 * DD);
        bn_stats<<<512, 256, 0, stream>>>(zagg, sums);
        float* dst = (l == LL - 1) ? out : zagg;
        bn_apply_relu<<<(NN * DD + 255) / 256, 256, 0, stream>>>(zagg, sums, dst);
        zin = zagg;
    }
}